// EGCN_21586505630270
// MI455X (gfx1250) — compile-verified
//
#include <hip/hip_runtime.h>
#include <math.h>

// ---- problem constants (match reference) ----
#define N_NODES 50000
#define N_EDGES 800000
#define NBASIS  10

typedef float v2f __attribute__((ext_vector_type(2)));
typedef float v8f __attribute__((ext_vector_type(8)));

static constexpr int   WAVES_PER_BLOCK = 4;
static constexpr int   THREADS         = WAVES_PER_BLOCK * 32;
static constexpr int   EDGES_PER_BLOCK = WAVES_PER_BLOCK * 16;
static constexpr int   XS_STRIDE       = 132;   // 128 + pad (banks)

static constexpr float SQRT3_F      = 1.7320508075688772f;
static constexpr float INV_SQRT3_F  = 0.5773502691896258f;
static constexpr float INV_SQRT10_F = 0.31622776601683794f;
static constexpr float EMB_SCALE_F  = 2.8234621998853827f;  // sqrt(10)/1.12
static constexpr float R_TO_BIN_F   = 4.5f;                 // 1/step, step = 2/9
static constexpr float W2_SCALE_F   = 0.025f;               // 0.1 (1/sqrt(100)) * 0.25 (1/sqrt(16))

__global__ __launch_bounds__(THREADS)
void egcn_fused_kernel(const float* __restrict__ pos,
                       const float* __restrict__ x,
                       const float* __restrict__ W1,
                       const float* __restrict__ W2,
                       const int*   __restrict__ esrc,
                       const int*   __restrict__ edst,
                       float*       __restrict__ out)
{
    __shared__ float W2s[100 * 128];                    // 0.025 * W2, row-major [k][n]
    __shared__ float W1s[NBASIS * 100];                 // [b][k]
    __shared__ float xsl[WAVES_PER_BLOCK][16 * XS_STRIDE];
    __shared__ float y1l[WAVES_PER_BLOCK][16][4];       // sqrt3 * unit vector

    const int tid  = threadIdx.x;
    const int wave = tid >> 5;
    const int lane = tid & 31;
    const int half = lane >> 4;     // which half-wave (K +0 / +2 selector)
    const int ml   = lane & 15;     // M for geometry, N-in-tile for WMMA B/C

    // ---- stage weights (fold GEMM scale and 1/sqrt(16) output scale into W2) ----
    for (int i = tid; i < 100 * 128; i += THREADS) W2s[i] = W2[i] * W2_SCALE_F;
    for (int i = tid; i < NBASIS * 100; i += THREADS) W1s[i] = W1[i];
    __syncthreads();

    // ---- per-edge geometry (each half-wave computes the same 16 edges) ----
    const int e    = blockIdx.x * EDGES_PER_BLOCK + wave * 16 + ml;
    const int srcL = esrc[e];
    const int dstL = edst[e];

    const float evx = pos[srcL * 3 + 0] - pos[dstL * 3 + 0];
    const float evy = pos[srcL * 3 + 1] - pos[dstL * 3 + 1];
    const float evz = pos[srcL * 3 + 2] - pos[dstL * 3 + 2];
    const float d2   = evx * evx + evy * evy + evz * evz + 1e-12f;
    const float rinv = __builtin_amdgcn_rsqf(d2);      // v_rsq_f32
    const float r    = d2 * rinv;                      // r = d2 / sqrt(d2)
    const float s3ir = SQRT3_F * rinv;
    if (half == 0) {
        y1l[wave][ml][0] = evx * s3ir;
        y1l[wave][ml][1] = evy * s3ir;
        y1l[wave][ml][2] = evz * s3ir;
    }

    // ---- radial basis ----
    float emb[NBASIS];
    const float rb = r * R_TO_BIN_F;
    #pragma unroll
    for (int b = 0; b < NBASIS; ++b) {
        const float d = rb - (float)b;
        emb[b] = __expf(-d * d) * EMB_SCALE_F;
    }

    // ---- h = silu(emb @ W1 / sqrt(10)); compute exactly the 50 values this
    //      lane contributes to A-fragments of v_wmma_f32_16x16x4_f32:
    //      k = 4*s + 2*half + t  (s=0..24, t=0..1).
    //      W1s reads are same-address broadcasts within each half-wave. ----
    float hreg[50];
    #pragma unroll
    for (int s = 0; s < 25; ++s) {
        #pragma unroll
        for (int t = 0; t < 2; ++t) {
            const int k = 4 * s + 2 * half + t;
            float acc = 0.0f;
            #pragma unroll
            for (int b = 0; b < NBASIS; ++b) acc += emb[b] * W1s[b * 100 + k];
            acc *= INV_SQRT10_F;
            // silu(t) = t * sigmoid(t), via v_exp_f32 + v_rcp_f32 (no IEEE divide chain)
            hreg[2 * s + t] = acc * __builtin_amdgcn_rcpf(1.0f + __expf(-acc));
        }
    }

    // ---- stage gathered x[src] rows for this wave's 16 edges ----
    float* xw = xsl[wave];
    for (int ee = 0; ee < 16; ++ee) {
        const int se = __shfl(srcL, ee, 32);
        const float* xr = x + (size_t)se * 128;
        #pragma unroll
        for (int q = lane; q < 128; q += 32) xw[ee * XS_STRIDE + q] = xr[q];
    }
    __syncthreads();

    // ---- w = h @ (0.025*W2) via native f32 WMMA, then scatter straight from
    //      the C-fragment layout: VGPR rr -> row M = rr + 8*half,
    //      lane -> column n = ml + 16*nt. Channel block c = nt>>1 is uniform. ----
    #pragma unroll
    for (int nt = 0; nt < 8; ++nt) {
        v8f c = {0.f, 0.f, 0.f, 0.f, 0.f, 0.f, 0.f, 0.f};
        const int ncol = nt * 16 + ml;
        #pragma unroll
        for (int s = 0; s < 25; ++s) {
            v2f a; a[0] = hreg[2 * s]; a[1] = hreg[2 * s + 1];
            const int krow = 4 * s + 2 * half;
            v2f b;
            b[0] = W2s[krow * 128 + ncol];
            b[1] = W2s[krow * 128 + 128 + ncol];
            c = __builtin_amdgcn_wmma_f32_16x16x4_f32(
                    false, a, false, b, (short)0, c, false, false);
        }

        const int cch = nt >> 1;               // compile-time after unroll
        const int u   = ml + 16 * (nt & 1);    // mul-channel index 0..31
        #pragma unroll
        for (int rr = 0; rr < 8; ++rr) {
            const int M  = rr + 8 * half;      // edge row of this C element
            const int dM = __shfl(dstL, M, 32);
            float* op = out + (size_t)dM * 256;
            const float wv = c[rr];            // scales pre-folded into W2s
            const float* xe = xw + M * XS_STRIDE;
            if (cch == 0) {
                // out0a = w0 * x0
                atomicAdd(op + u, wv * xe[u]);
            } else if (cch == 1) {
                // out1a = w1 * x0 * y1
                const float x0 = xe[u];
                atomicAdd(op + 64 + 3 * u + 0, wv * x0 * y1l[wave][M][0]);
                atomicAdd(op + 64 + 3 * u + 1, wv * x0 * y1l[wave][M][1]);
                atomicAdd(op + 64 + 3 * u + 2, wv * x0 * y1l[wave][M][2]);
            } else if (cch == 2) {
                // out1b = w2 * x1
                atomicAdd(op + 160 + 3 * u + 0, wv * xe[32 + 3 * u + 0]);
                atomicAdd(op + 160 + 3 * u + 1, wv * xe[32 + 3 * u + 1]);
                atomicAdd(op + 160 + 3 * u + 2, wv * xe[32 + 3 * u + 2]);
            } else {
                // out0b = w3 * (x1 . y1) / sqrt3
                const float d3 = xe[32 + 3 * u + 0] * y1l[wave][M][0]
                               + xe[32 + 3 * u + 1] * y1l[wave][M][1]
                               + xe[32 + 3 * u + 2] * y1l[wave][M][2];
                atomicAdd(op + 32 + u, wv * d3 * INV_SQRT3_F);
            }
        }
    }
}

extern "C" void kernel_launch(void* const* d_in, const int* in_sizes, int n_in,
                              void* d_out, int out_size, void* d_ws, size_t ws_size,
                              hipStream_t stream) {
    const float* pos = (const float*)d_in[0];
    const float* x   = (const float*)d_in[1];
    const float* W1  = (const float*)d_in[2];
    const float* W2  = (const float*)d_in[3];
    const int*   es  = (const int*)d_in[4];
    const int*   ed  = (const int*)d_in[5];
    float* out = (float*)d_out;

    // segment_sum target must start at zero (harness poisons d_out)
    hipMemsetAsync(out, 0, (size_t)out_size * sizeof(float), stream);

    const int grid = N_EDGES / EDGES_PER_BLOCK;   // 12500, exact
    egcn_fused_kernel<<<grid, THREADS, 0, stream>>>(pos, x, W1, W2, es, ed, out);
}